// GridEncoder_42528766165810
// MI455X (gfx1250) — compile-verified
//
#include <hip/hip_runtime.h>
#include <stdint.h>

// ============================================================================
// Instant-NGP hash-grid encoder for MI455X (gfx1250).
//
// Memory plan (from MI455X specs):
//   * Embedding table = 6,098,120 x float2 = 48.8 MB -> L2-resident (192 MB).
//   * 1M pts x 16 lvls x 8 corners x 8B = 1 GB of random float2 gathers: the
//     kernel is L2-gather bound; streaming I/O (~140 MB) is noise at 23.3 TB/s.
//   * Levels 0+1 (16,264 rows = 130,112 B) are cached in LDS via the Tensor
//     Data Mover (tensor_load_to_lds + s_wait_tensorcnt), absorbing 1/8 of all
//     gathers at LDS latency.
//   * 1024-thread workgroups: 130 KB LDS/block still allows 2 blocks/WGP
//     (260 KB <= 320 KB) -> 64 waves/WGP of latency hiding, and only
//     977 blocks x 130 KB ~= 127 MB of table-preload traffic (vs 0.5 GB with
//     256-thread blocks).
//   * Output (128 MB, write-once) uses non-temporal stores and input coords
//     non-temporal loads so the gather table stays hot in L2
//     (128 + 49 + 12 MB would otherwise thrash 192 MB).
//   * All per-level scales / resolutions / row offsets are compile-time
//     constants (match the reference's level_params()).
// ============================================================================

#define NUM_LEVEL 16
#define NUM_DENSE 5                  // levels 0..4 are collision-free (dense)
#define HASH_MASK ((1u << 19) - 1u)  // hashed levels: 2^19 entries
#define PRIME1 2654435761u
#define PRIME2 805459861u
#define LDS_ENTRIES 16264            // levels 0+1 rows cached in LDS (float2)
#define LDS_BYTES (LDS_ENTRIES * 8)  // 130112 B
#define BLOCK 1024

// scale_l = 16 * 1.3819^l - 1  (float32 of the python float64 values)
static constexpr float kScale[NUM_LEVEL] = {
    15.0f,        21.1104f,     29.5543618f,  41.2230725f,  57.3480638f,
    79.6311893f,  110.424240f,  152.977157f,  211.781035f,  293.042114f,
    405.336792f,  560.516813f,  774.960088f,  1071.29924f,  1480.81033f,
    2046.71369f};
static constexpr int kRes[NUM_DENSE] = {16, 23, 31, 43, 59};
static constexpr int kOff[NUM_LEVEL] = {
    0,       4096,    16264,   46056,   125568,
    330952,  855240,  1379528, 1903816, 2428104,
    2952392, 3476680, 4000968, 4525256, 5049544,
    5573832};

typedef unsigned int u32;
typedef u32   v4u __attribute__((ext_vector_type(4)));
typedef int   v4i __attribute__((ext_vector_type(4)));
typedef int   v8i __attribute__((ext_vector_type(8)));
typedef float v4f __attribute__((ext_vector_type(4)));  // for NT b128 stores

#if __has_builtin(__builtin_amdgcn_tensor_load_to_lds)
#define PRELOAD_TDM 1
#elif __has_builtin(__builtin_amdgcn_global_load_async_to_lds_b128)
#define PRELOAD_ASYNC 1
#endif

// Fill s_raw[0 .. LDS_BYTES) with embeddings rows [0, 16264) (levels 0+1).
__device__ __forceinline__ void preload_lds(const float* __restrict__ emb,
                                            float* __restrict__ s_raw) {
#if defined(PRELOAD_TDM)
  if (threadIdx.x < 32) {  // wave 0 only issues the DMA (uniform branch)
    typedef __attribute__((address_space(3))) float lds_f;
    u32 lds_off = (u32)(uint64_t)(lds_f*)s_raw;  // workgroup LDS byte offset
    uint64_t ga = (uint64_t)emb;
    // ---- D# group 0 (ISA 08 §8.3): count=1, lds_addr, global_addr, type=2
    v4u g0;
    g0.x = 1u;
    g0.y = lds_off;
    g0.z = (u32)ga;
    g0.w = ((u32)(ga >> 32) & 0x01FFFFFFu) | (2u << 30);
    // ---- D# group 1 (§8.4): data_size=3 (8B), 1 x 16264 tile of a 1 x 16264
    // tensor, dim0 stride = 16264 elements.
    v8i g1;
    g1[0] = (int)(3u << 16);                // workgroup_mask=0, data_size=8B
    g1[1] = (int)((u32)LDS_ENTRIES << 16);  // tensor_dim0[15:0] @ bits 63:48
    g1[2] = (int)(1u << 16);                // tensor_dim1 = 1
    g1[3] = (int)((u32)LDS_ENTRIES << 16);  // tile_dim0 @ bits 127:112
    g1[4] = 1;                              // tile_dim1 = 1
    g1[5] = LDS_ENTRIES;                    // tensor_dim0_stride[31:0]
    g1[6] = 0;
    g1[7] = 0;
    v4i z4 = {0, 0, 0, 0};                  // groups 2/3 unused (<=2D)
#if __clang_major__ >= 23
    v8i z8 = {0, 0, 0, 0, 0, 0, 0, 0};
    __builtin_amdgcn_tensor_load_to_lds(g0, g1, z4, z4, z8, 0);
#else
    __builtin_amdgcn_tensor_load_to_lds(g0, g1, z4, z4, 0);
#endif
    __builtin_amdgcn_s_wait_tensorcnt(0);
  }
#elif defined(PRELOAD_ASYNC)
  typedef __attribute__((address_space(1))) const char g_char;
  typedef __attribute__((address_space(3))) char l_char;
  g_char* gp = (g_char*)emb;
  l_char* lp = (l_char*)s_raw;
  const int chunks = LDS_BYTES / 16;  // 8132 x b128
  for (int i = threadIdx.x; i < chunks; i += blockDim.x)
    __builtin_amdgcn_global_load_async_to_lds_b128(gp + 16 * i, lp + 16 * i, 0, 0);
#if __has_builtin(__builtin_amdgcn_s_wait_asynccnt)
  __builtin_amdgcn_s_wait_asynccnt(0);
#else
  asm volatile("s_wait_asynccnt 0" ::: "memory");
#endif
#else
  {  // plain fallback
    const float2* g = (const float2*)emb;
    float2* s = (float2*)s_raw;
    for (int i = threadIdx.x; i < LDS_ENTRIES; i += blockDim.x) s[i] = g[i];
  }
#endif
  __syncthreads();
}

__global__ __launch_bounds__(BLOCK) void grid_encode_kernel(
    const float* __restrict__ xin, const float* __restrict__ emb,
    float* __restrict__ out, int n) {
  extern __shared__ __align__(16) float s_raw[];
  preload_lds(emb, s_raw);
  const float2* s_tab = (const float2*)s_raw;

  const int pt = blockIdx.x * BLOCK + threadIdx.x;
  if (pt >= n) return;

  // map [-1,1) -> [0,1); coords are read once -> non-temporal
  const float px = (__builtin_nontemporal_load(xin + 3 * pt + 0) + 1.0f) * 0.5f;
  const float py = (__builtin_nontemporal_load(xin + 3 * pt + 1) + 1.0f) * 0.5f;
  const float pz = (__builtin_nontemporal_load(xin + 3 * pt + 2) + 1.0f) * 0.5f;

  float res_out[2 * NUM_LEVEL];

  // ---------------- dense (collision-free) levels 0..4 ----------------
  // linear index = c0 + c1*R + c2*R^2 < R^3 <= table size, so no modulo.
  // Adjacent x-corners are contiguous float2 pairs -> compiler merges them
  // into b128 loads (ds_load_b128 / global_load_b128).
#pragma unroll
  for (int l = 0; l < NUM_DENSE; ++l) {
    const float s = kScale[l];
    const int R = kRes[l], R2 = R * R;
    const float fx = px * s, fy = py * s, fz = pz * s;
    const float gx = floorf(fx), gy = floorf(fy), gz = floorf(fz);
    const float tx = fx - gx, ty = fy - gy, tz = fz - gz;
    const float wx0 = 1.0f - tx, wy0 = 1.0f - ty, wz0 = 1.0f - tz;
    const int a0 = (int)gx, b0 = (int)gy * R, c0 = (int)gz * R2;
    const int a1 = a0 + 1, b1 = b0 + R, c1 = c0 + R2;
    const float2* tab =
        (l < 2) ? (s_tab + kOff[l]) : ((const float2*)emb + kOff[l]);
    float acc0 = 0.0f, acc1 = 0.0f;
#pragma unroll
    for (int k = 0; k < 8; ++k) {
      const int cx = (k >> 2) & 1, cy = (k >> 1) & 1, cz = k & 1;
      const float w = (cx ? tx : wx0) * (cy ? ty : wy0) * (cz ? tz : wz0);
      const int lin = (cx ? a1 : a0) + (cy ? b1 : b0) + (cz ? c1 : c0);
      const float2 f = tab[lin];  // LDS b64 for l<2, global b64 otherwise
      acc0 = fmaf(w, f.x, acc0);
      acc1 = fmaf(w, f.y, acc1);
    }
    res_out[2 * l + 0] = acc0;
    res_out[2 * l + 1] = acc1;
  }

  // ---------------- hashed levels 5..15 ----------------
  // unroll 2 -> 16 independent b64 gathers in flight per wait for MLP.
#pragma unroll 2
  for (int l = NUM_DENSE; l < NUM_LEVEL; ++l) {
    const float s = kScale[l];
    const float fx = px * s, fy = py * s, fz = pz * s;
    const float gx = floorf(fx), gy = floorf(fy), gz = floorf(fz);
    const float tx = fx - gx, ty = fy - gy, tz = fz - gz;
    const float wx0 = 1.0f - tx, wy0 = 1.0f - ty, wz0 = 1.0f - tz;
    const u32 ux = (u32)(int)gx, uy = (u32)(int)gy, uz = (u32)(int)gz;
    const u32 hx0 = ux,          hx1 = ux + 1u;
    const u32 hy0 = uy * PRIME1, hy1 = (uy + 1u) * PRIME1;
    const u32 hz0 = uz * PRIME2, hz1 = (uz + 1u) * PRIME2;
    const float2* tab = (const float2*)emb + kOff[l];
    float acc0 = 0.0f, acc1 = 0.0f;
#pragma unroll
    for (int k = 0; k < 8; ++k) {
      const int cx = (k >> 2) & 1, cy = (k >> 1) & 1, cz = k & 1;
      const float w = (cx ? tx : wx0) * (cy ? ty : wy0) * (cz ? tz : wz0);
      const u32 idx =
          ((cx ? hx1 : hx0) ^ (cy ? hy1 : hy0) ^ (cz ? hz1 : hz0)) & HASH_MASK;
      const float2 f = tab[idx];  // random b64 gather, L2-resident
      acc0 = fmaf(w, f.x, acc0);
      acc1 = fmaf(w, f.y, acc1);
    }
    res_out[2 * l + 0] = acc0;
    res_out[2 * l + 1] = acc1;
  }

  // 32 contiguous floats per point -> 8 x b128 non-temporal stores
  // (write-once stream; keep the gather table resident in L2).
  v4f* o = (v4f*)(out + (size_t)pt * 32);
#pragma unroll
  for (int i = 0; i < 8; ++i) {
    v4f v = {res_out[4 * i + 0], res_out[4 * i + 1],
             res_out[4 * i + 2], res_out[4 * i + 3]};
    __builtin_nontemporal_store(v, o + i);
  }
}

extern "C" void kernel_launch(void* const* d_in, const int* in_sizes, int n_in,
                              void* d_out, int out_size, void* d_ws,
                              size_t ws_size, hipStream_t stream) {
  const float* means = (const float*)d_in[0];  // [N,3] float32
  const float* emb = (const float*)d_in[1];    // [6098120,2] float32
  float* out = (float*)d_out;                  // [N,32] float32
  const int n = in_sizes[0] / 3;
  if (n <= 0) return;
  const int blocks = (n + BLOCK - 1) / BLOCK;
  grid_encode_kernel<<<blocks, BLOCK, LDS_BYTES, stream>>>(means, emb, out, n);
}